// per_pix_convolution_16269336117256
// MI455X (gfx1250) — compile-verified
//
#include <hip/hip_runtime.h>
#include <stdint.h>

#define N_IMG 8
#define C_CH  64
#define H_IMG 256
#define W_IMG 256
#define NTAP  9
#define TH    16
#define TW    64

typedef float        v4f __attribute__((ext_vector_type(4)));
typedef unsigned int v4u __attribute__((ext_vector_type(4)));
typedef unsigned int v8u __attribute__((ext_vector_type(8)));

__global__ __launch_bounds__(256)
void fac_per_pix_conv(const float* __restrict__ feature,
                      const float* __restrict__ attention,
                      float* __restrict__ out) {
  // Attention tile for this (n, h0, w0): 9 taps x 16 rows x 64 cols = 36 KB
  __shared__ float att_lds[NTAP * TH * TW];

  const int tid = threadIdx.x;
  const int w0  = blockIdx.x * TW;
  const int h0  = blockIdx.y * TH;
  const int n   = blockIdx.z;

  // ---- TDM: one tensor_load_to_lds pulls the whole 3D attention tile ----
  // Scalar (SCC) branch so only wave 0 reaches the TDM op (TDM ignores EXEC).
  if (__builtin_amdgcn_readfirstlane((int)threadIdx.x) == 0) {
    const uint64_t gaddr = (uint64_t)(uintptr_t)attention +
        4ull * ((uint64_t)n * NTAP * H_IMG * W_IMG +
                (uint64_t)h0 * W_IMG + (uint64_t)w0);
    const uint32_t laddr = (uint32_t)(uintptr_t)(void*)att_lds;

    // D# group 0: count=1 (valid, user mode), lds_addr, global_addr, type=2
    v4u g0;
    g0[0] = 1u;
    g0[1] = laddr;
    g0[2] = (uint32_t)gaddr;
    g0[3] = ((uint32_t)(gaddr >> 32) & 0x01FFFFFFu) | (2u << 30);

    // D# group 1: data_size=4B; tensor_dim0=W, tensor_dim1=H;
    // tile = 64 x 16 x 9; row stride = W, plane stride = H*W (in elements)
    v8u g1;
    g1[0] = 2u << 16;                                   // data_size = 4 bytes
    g1[1] = (uint32_t)W_IMG << 16;                      // tensor_dim0[15:0] @ bits 63:48
    g1[2] = (uint32_t)H_IMG << 16;                      // dim0 hi=0 | tensor_dim1[15:0]
    g1[3] = (uint32_t)TW << 16;                         // dim1 hi=0 | tile_dim0
    g1[4] = (uint32_t)TH | ((uint32_t)NTAP << 16);      // tile_dim1 | tile_dim2
    g1[5] = (uint32_t)W_IMG;                            // tensor_dim0_stride[31:0]
    g1[6] = 0u;                                         // stride0 hi | stride1[15:0] (=0)
    g1[7] = (uint32_t)((H_IMG * W_IMG) >> 16);          // tensor_dim1_stride[47:16]

    // D# group 2: tensor_dim2=9, dim2 stride = H*W, tile_dim3 unused
    v4u g2;
    g2[0] = NTAP;
    g2[1] = 0u;
    g2[2] = (uint32_t)(H_IMG * W_IMG);
    g2[3] = 0u;

    v4u g3;
    g3[0] = 0u; g3[1] = 0u; g3[2] = 0u; g3[3] = 0u;

    asm volatile("tensor_load_to_lds %0, %1, %2, %3\n\t"
                 "s_wait_tensorcnt 0x0"
                 :
                 : "s"(g0), "s"(g1), "s"(g2), "s"(g3)
                 : "memory");
  }
  __syncthreads();

  // ---- per-thread pixel assignment: one float4 of columns ----
  const int lx = (tid & 15) * 4;   // 0..60
  const int ly = tid >> 4;         // 0..15
  const int w  = w0 + lx;
  const int h  = h0 + ly;

  // Hoist the 9 per-pixel attention weights (x4 pixels) into VGPRs once;
  // they are channel-invariant.
  v4f A[NTAP];
#pragma unroll
  for (int t = 0; t < NTAP; ++t)
    A[t] = *(const v4f*)&att_lds[(t * TH + ly) * TW + lx];

  const bool has_l = (w > 0);
  const bool has_r = (w + 4 < W_IMG);

  const size_t plane = (size_t)H_IMG * W_IMG;
  const float* f = feature + ((size_t)n * C_CH) * plane + (size_t)h * W_IMG + w;
  float*       o = out     + ((size_t)n * C_CH) * plane + (size_t)h * W_IMG + w;

#pragma unroll 2
  for (int c = 0; c < C_CH; ++c, f += plane, o += plane) {
    v4f acc = (v4f){0.f, 0.f, 0.f, 0.f};
#pragma unroll
    for (int dy = -1; dy <= 1; ++dy) {
      v4f   m   = (v4f){0.f, 0.f, 0.f, 0.f};
      float lft = 0.f, rgt = 0.f;
      const int hh = h + dy;
      if ((unsigned)hh < (unsigned)H_IMG) {      // zero-pad rows outside image
        const float* row = f + dy * W_IMG;
        m = *(const v4f*)row;                    // global_load_b128, L1/L2 reuse
        if (has_l) lft = row[-1];
        if (has_r) rgt = row[4];
      }
      if (dy == 0) acc += m;                     // residual term
      const int tb = (dy + 1) * 3;
      acc.x += A[tb + 0].x * lft + A[tb + 1].x * m.x + A[tb + 2].x * m.y;
      acc.y += A[tb + 0].y * m.x + A[tb + 1].y * m.y + A[tb + 2].y * m.z;
      acc.z += A[tb + 0].z * m.y + A[tb + 1].z * m.z + A[tb + 2].z * m.w;
      acc.w += A[tb + 0].w * m.z + A[tb + 1].w * m.w + A[tb + 2].w * rgt;
    }
    // Output is streamed once and never re-read: non-temporal store keeps
    // L2 (192 MB) resident with feature+attention (~153 MB) instead.
    __builtin_nontemporal_store(acc, (v4f*)o);
  }
}

extern "C" void kernel_launch(void* const* d_in, const int* in_sizes, int n_in,
                              void* d_out, int out_size, void* d_ws, size_t ws_size,
                              hipStream_t stream) {
  (void)in_sizes; (void)n_in; (void)out_size; (void)d_ws; (void)ws_size;
  const float* feature   = (const float*)d_in[0];
  const float* attention = (const float*)d_in[1];
  float* out = (float*)d_out;

  dim3 grid(W_IMG / TW, H_IMG / TH, N_IMG);   // 4 x 16 x 8 = 512 blocks
  dim3 block(256);                            // 8 wave32 waves
  fac_per_pix_conv<<<grid, block, 0, stream>>>(feature, attention, out);
}